// OversizeConv2d_8693013807253
// MI455X (gfx1250) — compile-verified
//
#include <hip/hip_runtime.h>
#include <hip/hip_bf16.h>

// ---------------------------------------------------------------------------
// OversizeConv2d on MI455X (gfx1250):
// separable depthwise conv with kernel length 2H-1 and pad H-1 degenerates to
// two dense 64x64 matmuls per (batch, channel):
//     Y_bc = Mh_c * X_bc * MwT_c + (h_bias_c * Sw[l] + w_bias_c)
// Implemented with V_WMMA_F32_16X16X4_F32 (fp32 matrix engine, wave32).
// All WMMA operand K-pairs are stored contiguously (8B aligned) so each
// operand is a single ds_load_b64 / global_load_b64 -> no repacking movs.
// ---------------------------------------------------------------------------

typedef __attribute__((ext_vector_type(2))) float v2f;
typedef __attribute__((ext_vector_type(8))) float v8f;

#define DIM   128   // channels
#define BATCH 32
#define HW    64    // H == W == 64
#define KTAP  13
#define KFULL 127   // 2*HW - 1
#define PITCH 68    // LDS row pitch: even (8B align) + conflict-free bank split

// ---------------------------------------------------------------------------
// Pass A: build per-channel Toeplitz operands + folded bias in workspace.
//   Mh [c][i][j]     = hk_c[j - i + 63]                  (row-major 64x64)
//   Bpk[c][p][n]     = { wk_c[2p - n + 63],
//                        wk_c[2p+1 - n + 63] }           (K-pair packed float2)
//   bias2[c][l]      = h_bias_c * Sw_c[l] + w_bias_c,  Sw[l] = sum_j wk[j-l+63]
// hk/wk are the 13->127 linear interpolations (align_corners=True).
// ---------------------------------------------------------------------------
__global__ __launch_bounds__(256) void build_mats(
    const float* __restrict__ hw, const float* __restrict__ ww,
    const float* __restrict__ hb, const float* __restrict__ wb,
    float* __restrict__ Mh, float2* __restrict__ Bpk, float* __restrict__ bias2)
{
    __shared__ float hk[KFULL];
    __shared__ float wk[KFULL];
    const int c = blockIdx.x;
    const int t = threadIdx.x;

    if (t < KFULL) {
        const float scale = (float)(KTAP - 1) / (float)(KFULL - 1); // 12/126
        float pos  = (float)t * scale;
        int   lo   = (int)floorf(pos);
        int   hi   = min(lo + 1, KTAP - 1);
        float frac = pos - (float)lo;
        hk[t] = hw[c * KTAP + lo] * (1.0f - frac) + hw[c * KTAP + hi] * frac;
        wk[t] = ww[c * KTAP + lo] * (1.0f - frac) + ww[c * KTAP + hi] * frac;
    }
    __syncthreads();

    if (t < HW) {
        float s = 0.0f;
        #pragma unroll 8
        for (int j = 0; j < HW; ++j) s += wk[j - t + (HW - 1)];
        bias2[c * HW + t] = hb[c] * s + wb[c];
    }

    // Mh: row-major Toeplitz for the H pass
    for (int idx = t; idx < HW * HW; idx += 256) {
        int i = idx >> 6;
        int j = idx & 63;
        Mh[c * HW * HW + idx] = hk[j - i + (HW - 1)];
    }
    // Bpk: K-pair interleaved Toeplitz for the W pass (B operand of GEMM2)
    for (int idx = t; idx < (HW / 2) * HW; idx += 256) {
        int p = idx >> 6;       // k-pair index 0..31
        int n = idx & 63;       // output column
        float2 v;
        v.x = wk[(2 * p)     - n + (HW - 1)];
        v.y = wk[(2 * p + 1) - n + (HW - 1)];
        Bpk[c * (HW / 2) * HW + idx] = v;
    }
}

// ---------------------------------------------------------------------------
// Pass B: one block (4 wave32) per (b,c) slab.
//   Wave w owns output rows [16w, 16w+16).
//   GEMM1: T = Mh * X   (A: global b64 panel, B: ds_load_b64 from XsT)
//   GEMM2: Y = T * MwT  (A: ds_load_b64 from Ts, B: global b64 from Bpk) + bias
// WMMA f32 16x16x4 lane layout:
//   A: lane -> m = lane&15, k-pair base kb = (lane>>4)*2, VGPRs {k=kb, k=kb+1}
//   C/D: VGPR r -> m = r + 8*(lane>>4), n = lane&15
// ---------------------------------------------------------------------------
__global__ __launch_bounds__(128) void conv_gemm(
    const float* __restrict__ x,
    const float* __restrict__ Mh, const float2* __restrict__ Bpk,
    const float* __restrict__ bias2, float* __restrict__ y)
{
    __shared__ float XsT[HW * PITCH];   // X transposed: XsT[n*PITCH + k]
    __shared__ float Ts [HW * PITCH];   // T row-major : Ts [m*PITCH + k]

    const int s = blockIdx.x;          // s = b*DIM + c
    const int c = s & (DIM - 1);
    const float* __restrict__ xb = x + (size_t)s * (HW * HW);

    const int t    = threadIdx.x;
    const int wave = t >> 5;
    const int lane = t & 31;
    const int m16  = lane & 15;
    const int kb   = (lane >> 4) << 1;     // 0 or 2
    const int mhi  = (lane >> 4) << 3;     // 0 or 8
    const int r0   = wave << 4;            // wave's row stripe

    // ---- stage X_bc into LDS transposed (coalesced global reads) ----
    for (int i = t; i < HW * HW; i += 128) {
        int r = i >> 6;      // H row  (K dim of GEMM1)
        int n = i & 63;      // W col  (N dim of GEMM1)
        XsT[n * PITCH + r] = xb[i];
    }
    __syncthreads();

    const float*  __restrict__ MhC = Mh  + (size_t)c * (HW * HW);
    const float2* __restrict__ BpC = Bpk + (size_t)c * ((HW / 2) * HW);

    // ---- GEMM1: T = Mh * X ----
    v2f Ap[16];
    #pragma unroll
    for (int kk = 0; kk < 16; ++kk) {
        // 8B-aligned global load of the lane's A K-pair (L2-resident matrix)
        Ap[kk] = *(const v2f*)(MhC + (r0 + m16) * HW + (kk << 2) + kb);
    }
    v8f acc[4];
    #pragma unroll
    for (int nt = 0; nt < 4; ++nt) acc[nt] = (v8f){0, 0, 0, 0, 0, 0, 0, 0};

    #pragma unroll
    for (int nt = 0; nt < 4; ++nt) {
        const int n = (nt << 4) + m16;
        #pragma unroll
        for (int kk = 0; kk < 16; ++kk) {
            // single ds_load_b64: {B[k][n], B[k+1][n]} adjacent in XsT
            v2f Bv = *(const v2f*)(XsT + n * PITCH + (kk << 2) + kb);
            acc[nt] = __builtin_amdgcn_wmma_f32_16x16x4_f32(
                false, Ap[kk], false, Bv, (short)0, acc[nt], false, false);
        }
    }

    // ---- spill T stripe to LDS (row-major, conflict-free banks) ----
    #pragma unroll
    for (int nt = 0; nt < 4; ++nt) {
        #pragma unroll
        for (int r = 0; r < 8; ++r) {
            Ts[(r0 + r + mhi) * PITCH + (nt << 4) + m16] = acc[nt][r];
        }
    }
    __syncthreads();

    // ---- GEMM2: Y = T * MwT ----
    #pragma unroll
    for (int kk = 0; kk < 16; ++kk) {
        // single ds_load_b64 of the lane's A K-pair
        Ap[kk] = *(const v2f*)(Ts + (r0 + m16) * PITCH + (kk << 2) + kb);
    }
    #pragma unroll
    for (int nt = 0; nt < 4; ++nt) acc[nt] = (v8f){0, 0, 0, 0, 0, 0, 0, 0};

    #pragma unroll
    for (int nt = 0; nt < 4; ++nt) {
        const int n = (nt << 4) + m16;
        #pragma unroll
        for (int kk = 0; kk < 16; ++kk) {
            // single global_load_b64 of the pre-packed B K-pair (L2-resident)
            const int p = (kk << 1) + (kb >> 1);   // k-pair index
            v2f Bv = *(const v2f*)(BpC + (p << 6) + n);
            acc[nt] = __builtin_amdgcn_wmma_f32_16x16x4_f32(
                false, Ap[kk], false, Bv, (short)0, acc[nt], false, false);
        }
    }

    // ---- add folded bias, store (64B-contiguous per half-wave row) ----
    float* __restrict__ yb = y + (size_t)s * (HW * HW);
    #pragma unroll
    for (int nt = 0; nt < 4; ++nt) {
        const int n = (nt << 4) + m16;
        const float bval = bias2[c * HW + n];
        #pragma unroll
        for (int r = 0; r < 8; ++r) {
            yb[(r0 + r + mhi) * HW + n] = acc[nt][r] + bval;
        }
    }
}

// ---------------------------------------------------------------------------
extern "C" void kernel_launch(void* const* d_in, const int* in_sizes, int n_in,
                              void* d_out, int out_size, void* d_ws, size_t ws_size,
                              hipStream_t stream) {
    const float* x  = (const float*)d_in[0];   // (32,128,64,64)
    const float* hw = (const float*)d_in[1];   // (128,1,13,1)
    const float* ww = (const float*)d_in[2];   // (128,1,1,13)
    const float* hb = (const float*)d_in[3];   // (128,)
    const float* wb = (const float*)d_in[4];   // (128,)
    float* y = (float*)d_out;

    float*  Mh    = (float*)d_ws;                        // 128*64*64 floats = 2 MB
    float2* Bpk   = (float2*)(Mh + DIM * HW * HW);       // 128*32*64 float2 = 2 MB
    float*  bias2 = (float*)(Bpk + DIM * (HW / 2) * HW); // 32 KB

    build_mats<<<DIM, 256, 0, stream>>>(hw, ww, hb, wb, Mh, Bpk, bias2);
    conv_gemm<<<BATCH * DIM, 128, 0, stream>>>(x, Mh, Bpk, bias2, y);
}